// ChamferDistanceLoss_68272800137445
// MI455X (gfx1250) — compile-verified
//
#include <hip/hip_runtime.h>
#include <hip/hip_bf16.h>

typedef float v2f __attribute__((ext_vector_type(2)));
typedef float v8f __attribute__((ext_vector_type(8)));

#define BATCH 8
#define NPTS 4096
#define CHUNK 512
#define ROWTILES 32  // NPTS / 128 row-tiles per batch

// One block: 256 threads = 8 wave32; each wave owns a 16-row strip of a
// 128-row tile and sweeps all 4096 "column" points, staged in LDS in WMMA
// B-operand layout. Both chamfer directions run in one launch (grid doubled):
// direction 0 computes dist_min1 (rows = points1), direction 1 computes
// dist_min2 (rows = points2) via the transposed product.
//
// V_WMMA_F32_16X16X4_F32 with norm folding:
//   A row i = [xr, yr, zr, 1]                   (fixed per wave)
//   B col j = [-2xc, -2yc, -2zc, ||c_j||^2]     (from LDS, one b64 per lane)
//   D       = -2*dot + ||c_j||^2
//   dist    = D + ||r_i||^2  -> row-constant, added once in the epilogue,
//                               so the inner loop is just min(racc, D).
__global__ __launch_bounds__(256) void chamfer_rowmin_kernel(
    const float* __restrict__ p1, const float* __restrict__ p2,
    float* __restrict__ min1, float* __restrict__ min2) {
    __shared__ v2f sB[2 * CHUNK];  // [half][j]: half0=(-2x,-2y), half1=(-2z,norm)

    int blk = blockIdx.x;
    const float* rowsPts;
    const float* colsPts;
    float* outMin;
    if (blk < BATCH * ROWTILES) {
        rowsPts = p1; colsPts = p2; outMin = min1;
    } else {
        blk -= BATCH * ROWTILES;
        rowsPts = p2; colsPts = p1; outMin = min2;
    }

    const int tid = threadIdx.x;
    const int lane = tid & 31;
    const int wave = tid >> 5;
    const int b = blk >> 5;
    const int i0 = (blk & 31) * 128;

    // ---- A operand (16x4 f32): lanes 0-15 -> (K0,K1)=(x,y); 16-31 -> (K2,K3)=(z,1)
    const int iLocal = lane & 15;
    const int i = i0 + wave * 16 + iLocal;
    const float* pa = rowsPts + ((size_t)b * NPTS + i) * 3;
    const float ax = pa[0], ay = pa[1], az = pa[2];
    const float ni = ax * ax + ay * ay + az * az;
    v2f A;
    A.x = (lane < 16) ? ax : az;
    A.y = (lane < 16) ? ay : 1.0f;

    // ---- per-lane ||row||^2 for this lane's 8 C-rows (M = v + 8*(lane>=16));
    // identical across each 16-lane half, so it can be added post-reduction.
    const int nbase = (lane & 16) ? 8 : 0;
    v8f an;
#pragma unroll
    for (int v = 0; v < 8; ++v) an[v] = __shfl(ni, nbase + v, 32);

    v8f racc;
#pragma unroll
    for (int v = 0; v < 8; ++v) racc[v] = __builtin_inff();

    // per-lane B base: half-wave selects which K-pair it supplies
    const v2f* sBl = sB + ((lane & 16) ? CHUNK : 0);

    for (int j0 = 0; j0 < NPTS; j0 += CHUNK) {
        // stage CHUNK column-points pre-packed in WMMA B layout
        for (int k = tid; k < CHUNK; k += 256) {
            const float* pb = colsPts + ((size_t)b * NPTS + j0 + k) * 3;
            float x = pb[0], y = pb[1], z = pb[2];
            v2f h0, h1;
            h0.x = -2.0f * x;
            h0.y = -2.0f * y;
            h1.x = -2.0f * z;
            h1.y = x * x + y * y + z * z;
            sB[k] = h0;
            sB[CHUNK + k] = h1;
        }
        __syncthreads();

#pragma unroll 4
        for (int jt = 0; jt < CHUNK / 16; ++jt) {
            v2f B = sBl[jt * 16 + iLocal];  // one ds_load_b64
            v8f c = {};
            v8f d = __builtin_amdgcn_wmma_f32_16x16x4_f32(
                /*neg_a=*/false, A, /*neg_b=*/false, B,
                /*c_mod=*/(short)0, c, /*reuse_a=*/false, /*reuse_b=*/false);
#pragma unroll
            for (int v = 0; v < 8; ++v) racc[v] = fminf(racc[v], d[v]);
        }
        __syncthreads();
    }

    // ---- epilogue: add row norms, then cross-lane min over the 16 lanes/row
#pragma unroll
    for (int v = 0; v < 8; ++v) racc[v] += an[v];
#pragma unroll
    for (int s = 1; s <= 8; s <<= 1) {
#pragma unroll
        for (int v = 0; v < 8; ++v)
            racc[v] = fminf(racc[v], __shfl_xor(racc[v], s, 32));
    }
    if (iLocal == 0) {
        const int rb = i0 + wave * 16 + nbase;
#pragma unroll
        for (int v = 0; v < 8; ++v)
            outMin[(size_t)b * NPTS + rb + v] = racc[v];
    }
}

__global__ __launch_bounds__(256) void chamfer_reduce_kernel(
    const float* __restrict__ min1, const float* __restrict__ min2,
    float* __restrict__ out) {
    __shared__ float s1[256], s2[256];
    const int tid = threadIdx.x;
    float a = 0.0f, c = 0.0f;
    for (int k = tid; k < BATCH * NPTS; k += 256) {
        a += min1[k];
        c += min2[k];
    }
    s1[tid] = a;
    s2[tid] = c;
    __syncthreads();
    for (int s = 128; s > 0; s >>= 1) {
        if (tid < s) {
            s1[tid] += s1[tid + s];
            s2[tid] += s2[tid + s];
        }
        __syncthreads();
    }
    if (tid == 0) {
        const float inv = 1.0f / (float)(BATCH * NPTS);
        float loss1 = s1[0] * inv;  // mean of dist_min1
        float loss2 = s2[0] * inv;  // mean of dist_min2
        out[0] = (loss1 + loss2) / (float)BATCH;
    }
}

extern "C" void kernel_launch(void* const* d_in, const int* in_sizes, int n_in,
                              void* d_out, int out_size, void* d_ws, size_t ws_size,
                              hipStream_t stream) {
    (void)in_sizes; (void)n_in; (void)out_size; (void)ws_size;
    const float* p1 = (const float*)d_in[0];
    const float* p2 = (const float*)d_in[1];
    float* out = (float*)d_out;

    float* min1 = (float*)d_ws;                          // B*N floats
    float* min2 = min1 + (size_t)BATCH * NPTS;           // B*M floats

    chamfer_rowmin_kernel<<<2 * BATCH * ROWTILES, 256, 0, stream>>>(p1, p2, min1, min2);
    chamfer_reduce_kernel<<<1, 256, 0, stream>>>(min1, min2, out);
}